// AttentionWrapper_19138374271641
// MI455X (gfx1250) — compile-verified
//
#include <hip/hip_runtime.h>
#include <cstdint>

// Problem constants (match reference)
#define BB     64
#define TT     4096
#define DIMN   256
#define NFILT  32
#define KSIZE  31
#define PADW   15
#define NEGV   (-1e30f)

#define TILE_T  256            // timesteps per score workgroup
#define WC_K    32             // K padded 31 -> 32 (row 31 zeroed)
#define W2S     260            // LDS row pitch (in float2) for paired Wc rows

typedef __attribute__((ext_vector_type(2))) float v2f;
typedef __attribute__((ext_vector_type(8))) float v8f;

// Branch-free fast tanh: 1 - 2/(exp(2x)+1). Saturates correctly at +/-1.
__device__ __forceinline__ float tanh_fast(float x)
{
    const float e = __expf(2.0f * x);               // v_exp_f32
    return 1.0f - 2.0f * __builtin_amdgcn_rcpf(e + 1.0f);
}

// ---------------------------------------------------------------------------
// Kernel 1: qproj[b][d] = sum_e query[b][e]*Wq[d][e]
//           Wc[k][d]    = sum_f Wloc[d][f]*conv_w[f][0][k]   (k==31 -> 0)
// ---------------------------------------------------------------------------
__global__ __launch_bounds__(256)
void prep_kernel(const float* __restrict__ query,
                 const float* __restrict__ Wq,
                 const float* __restrict__ Wloc,
                 const float* __restrict__ conv_w,
                 float* __restrict__ qproj,
                 float* __restrict__ Wc)
{
    __shared__ float qs[DIMN];
    const int tid = threadIdx.x;
    if ((int)blockIdx.x < BB) {
        const int b = blockIdx.x;
        qs[tid] = query[b * DIMN + tid];
        __syncthreads();
        const float4* wrow = reinterpret_cast<const float4*>(Wq + (size_t)tid * DIMN);
        float s = 0.f;
#pragma unroll 4
        for (int i = 0; i < DIMN / 4; ++i) {
            float4 w = wrow[i];
            s += qs[4*i+0]*w.x + qs[4*i+1]*w.y + qs[4*i+2]*w.z + qs[4*i+3]*w.w;
        }
        qproj[b * DIMN + tid] = s;
    } else {
        for (int idx = tid; idx < WC_K * DIMN; idx += 256) {
            const int k = idx >> 8;
            const int d = idx & (DIMN - 1);
            float s = 0.f;
            if (k < KSIZE) {
#pragma unroll 8
                for (int f = 0; f < NFILT; ++f)
                    s += Wloc[d * NFILT + f] * conv_w[f * KSIZE + k];
            }
            Wc[k * DIMN + d] = s;   // row-major [k][d]
        }
    }
}

// ---------------------------------------------------------------------------
// Kernel 2: fused loc-GEMM (WMMA f32 16x16x4) + keys + tanh score + mask.
// Grid: (T/TILE_T, B). Block: 256 threads = 8 waves; wave owns 2 M-tiles.
// N-tile outer loop: B fragments double-buffered through LDS, A fragments
// and per-column qproj/v_w values fully register-resident.
// ---------------------------------------------------------------------------
__global__ __launch_bounds__(256)
void score_kernel(const float*   __restrict__ keys,
                  const float*   __restrict__ prev_alignment,
                  const uint8_t* __restrict__ mask,     // jnp bool -> 1 byte/elem
                  const float*   __restrict__ v_w,
                  const float*   __restrict__ v_b,
                  const float*   __restrict__ qproj,
                  const float*   __restrict__ Wc,
                  float*         __restrict__ align_raw)
{
    __shared__ float  pa_win[TILE_T + 32];     // sliding window of prev_alignment
    __shared__ float2 wc2[16 * W2S];           // paired rows: wc2[j][d] = (Wc[2j][d], Wc[2j+1][d])
    __shared__ float  qp_lds[DIMN];
    __shared__ float  vw_lds[DIMN];

    const int b    = blockIdx.y;
    const int t0   = blockIdx.x * TILE_T;
    const int tid  = threadIdx.x;
    const int lane = tid & 31;
    const int wave = tid >> 5;
    const int half = lane >> 4;     // 0: lanes 0-15, 1: lanes 16-31
    const int l16  = lane & 15;

    // ---- stage LDS ----
    qp_lds[tid] = qproj[b * DIMN + tid];
    vw_lds[tid] = v_w[tid];
    for (int idx = tid; idx < 16 * DIMN; idx += 256) {
        const int j = idx >> 8;
        const int d = idx & (DIMN - 1);
        wc2[j * W2S + d] = make_float2(Wc[(2*j) * DIMN + d], Wc[(2*j+1) * DIMN + d]);
    }
    for (int j = tid; j < TILE_T + 32; j += 256) {
        const int t = t0 + j - PADW;
        pa_win[j] = (t >= 0 && t < TT) ? prev_alignment[(size_t)b * TT + t] : 0.f;
    }
    __syncthreads();

    const float vb = v_b[0];

    // ---- hoist per-column qproj / v_w values into registers ----
    float qvr[DIMN / 16], vwr[DIMN / 16];
#pragma unroll
    for (int nt = 0; nt < DIMN / 16; ++nt) {
        qvr[nt] = qp_lds[nt * 16 + l16];
        vwr[nt] = vw_lds[nt * 16 + l16];
    }

    // ---- hoist A fragments (pa window) for this wave's 2 M-tiles ----
    // f32 16x16x4 A layout: lanes 0-15 hold K=0,1 ; lanes 16-31 hold K=2,3.
    v2f afrag[2][8];
#pragma unroll
    for (int mi = 0; mi < 2; ++mi) {
#pragma unroll
        for (int kk = 0; kk < 8; ++kk) {
            const int base = (wave * 2 + mi) * 16 + l16 + 4 * kk + 2 * half;
            afrag[mi][kk].x = pa_win[base];
            afrag[mi][kk].y = pa_win[base + 1];
        }
    }

    float sacc[2][8];
#pragma unroll
    for (int mi = 0; mi < 2; ++mi)
#pragma unroll
        for (int i = 0; i < 8; ++i) sacc[mi][i] = 0.f;

    // ---- B fragment double buffer (depends only on nt) ----
    v2f wf[8];
#pragma unroll
    for (int kk = 0; kk < 8; ++kk) {
        const float2 t = wc2[(2 * kk + half) * W2S + l16];   // nt = 0
        wf[kk].x = t.x; wf[kk].y = t.y;
    }

    for (int nt = 0; nt < DIMN / 16; ++nt) {
        v2f wn[8];
        if (nt < DIMN / 16 - 1) {
            const int ncol = (nt + 1) * 16 + l16;
#pragma unroll
            for (int kk = 0; kk < 8; ++kk) {
                const float2 t = wc2[(2 * kk + half) * W2S + ncol];
                wn[kk].x = t.x; wn[kk].y = t.y;
            }
        }

        const int   col = nt * 16 + l16;
        const float qv  = qvr[nt];
        const float vw  = vwr[nt];

#pragma unroll
        for (int mi = 0; mi < 2; ++mi) {
            v8f acc = {};
#pragma unroll
            for (int kk = 0; kk < 8; ++kk) {
                acc = __builtin_amdgcn_wmma_f32_16x16x4_f32(
                        false, afrag[mi][kk], false, wf[kk],
                        (short)0, acc, false, false);
            }

            // D layout: lane<16 -> (M=i, N=l16); lane>=16 -> (M=i+8, N=l16)
            const float* krow = keys
                + ((size_t)b * TT + t0 + (wave * 2 + mi) * 16 + 8 * half) * DIMN + col;
            float kv[8];
#pragma unroll
            for (int i = 0; i < 8; ++i) kv[i] = krow[(size_t)i * DIMN];
#pragma unroll
            for (int i = 0; i < 8; ++i) {
                const float x = acc[i] + qv + kv[i];
                sacc[mi][i] += tanh_fast(x) * vw;
            }
        }

#pragma unroll
        for (int kk = 0; kk < 8; ++kk) wf[kk] = wn[kk];
    }

    // ---- reduce over the 16 columns held in each 16-lane half ----
#pragma unroll
    for (int off = 8; off >= 1; off >>= 1) {
#pragma unroll
        for (int mi = 0; mi < 2; ++mi)
#pragma unroll
            for (int i = 0; i < 8; ++i)
                sacc[mi][i] += __shfl_xor(sacc[mi][i], off, 32);
    }

    if (l16 == 0) {
#pragma unroll
        for (int mi = 0; mi < 2; ++mi) {
#pragma unroll
            for (int i = 0; i < 8; ++i) {
                const int   t = t0 + (wave * 2 + mi) * 16 + i + 8 * half;
                const float s = sacc[mi][i] + vb;
                align_raw[(size_t)b * TT + t] = mask[(size_t)b * TT + t] ? NEGV : s;
            }
        }
    }
}

// ---------------------------------------------------------------------------
// Kernel 3: per-batch softmax (in place on align) + context = align @ values
// Grid: (B). Block: 1024 threads.
// ---------------------------------------------------------------------------
__global__ __launch_bounds__(1024)
void softmax_ctx_kernel(const float* __restrict__ values,
                        float*       __restrict__ align_io,
                        float*       __restrict__ attention)
{
    __shared__ float a_lds[TT];          // 16 KB: exp / normalized weights
    __shared__ float part[16 * DIMN];    // 16 KB: context partials
    __shared__ float red[32];

    const int b    = blockIdx.x;
    const int tid  = threadIdx.x;
    const int lane = tid & 31;
    const int wid  = tid >> 5;

    // ---- load scores, row max ----
    float sl[4];
    float m = -__builtin_inff();
#pragma unroll
    for (int j = 0; j < 4; ++j) {
        const int t = tid + j * 1024;
        const float s = align_io[(size_t)b * TT + t];
        sl[j] = s;
        m = fmaxf(m, s);
    }
#pragma unroll
    for (int off = 16; off >= 1; off >>= 1) m = fmaxf(m, __shfl_xor(m, off, 32));
    if (lane == 0) red[wid] = m;
    __syncthreads();
    if (tid < 32) {
        float v = red[tid];
#pragma unroll
        for (int off = 16; off >= 1; off >>= 1) v = fmaxf(v, __shfl_xor(v, off, 32));
        if (tid == 0) red[0] = v;
    }
    __syncthreads();
    const float gmax = red[0];
    __syncthreads();   // protect red[] before reuse

    // ---- exp + sum ----
    float es = 0.f;
#pragma unroll
    for (int j = 0; j < 4; ++j) {
        const int t = tid + j * 1024;
        const float e = __expf(sl[j] - gmax);
        a_lds[t] = e;
        es += e;
    }
#pragma unroll
    for (int off = 16; off >= 1; off >>= 1) es += __shfl_xor(es, off, 32);
    if (lane == 0) red[wid] = es;
    __syncthreads();
    if (tid < 32) {
        float v = red[tid];
#pragma unroll
        for (int off = 16; off >= 1; off >>= 1) v += __shfl_xor(v, off, 32);
        if (tid == 0) red[0] = v;
    }
    __syncthreads();
    const float inv = 1.0f / red[0];

    // ---- normalize (LDS + global out) ----
#pragma unroll
    for (int j = 0; j < 4; ++j) {
        const int t = tid + j * 1024;
        const float an = a_lds[t] * inv;
        a_lds[t] = an;
        align_io[(size_t)b * TT + t] = an;
    }
    __syncthreads();

    // ---- context: 16 t-groups x 64 threads, float4 over d ----
    const int g = tid >> 6;            // 0..15
    const int q = tid & 63;            // float4 column
    const float4* vrow = reinterpret_cast<const float4*>(values)
                       + (size_t)b * TT * (DIMN / 4) + q;
    float4 acc = make_float4(0.f, 0.f, 0.f, 0.f);
    for (int t = g; t < TT; t += 16) {
        if (t + 16 < TT)
            __builtin_prefetch(&vrow[(size_t)(t + 16) * (DIMN / 4)], 0, 0);
        const float  a = a_lds[t];
        const float4 v = vrow[(size_t)t * (DIMN / 4)];
        acc.x += a * v.x; acc.y += a * v.y; acc.z += a * v.z; acc.w += a * v.w;
    }
    reinterpret_cast<float4*>(part)[g * 64 + q] = acc;
    __syncthreads();
    if (tid < DIMN) {
        float s = 0.f;
#pragma unroll
        for (int g2 = 0; g2 < 16; ++g2) s += part[g2 * DIMN + tid];
        attention[(size_t)b * DIMN + tid] = s;
    }
}

// ---------------------------------------------------------------------------
extern "C" void kernel_launch(void* const* d_in, const int* in_sizes, int n_in,
                              void* d_out, int out_size, void* d_ws, size_t ws_size,
                              hipStream_t stream)
{
    (void)in_sizes; (void)n_in; (void)out_size; (void)ws_size;

    const float*   query    = (const float*)  d_in[0];   // (B, DIM)
    const float*   keys     = (const float*)  d_in[1];   // (B, T, DIM)
    const float*   values   = (const float*)  d_in[2];   // (B, T, DIM)
    const float*   prev_al  = (const float*)  d_in[3];   // (B, T)
    const uint8_t* mask     = (const uint8_t*)d_in[4];   // (B, T) bool
    const float*   Wq       = (const float*)  d_in[5];   // (DIM, DIM)
    const float*   conv_w   = (const float*)  d_in[6];   // (NFILT, 1, KSIZE)
    const float*   Wloc     = (const float*)  d_in[7];   // (DIM, NFILT)
    const float*   v_w      = (const float*)  d_in[8];   // (1, DIM)
    const float*   v_b      = (const float*)  d_in[9];   // (1,)

    float* att_out   = (float*)d_out;                    // (B, DIM)
    float* align_out = att_out + (size_t)BB * DIMN;      // (B, T)

    float* qproj_ws = (float*)d_ws;                      // B*DIM floats
    float* wc_ws    = qproj_ws + (size_t)BB * DIMN;      // WC_K*DIM floats

    prep_kernel<<<dim3(BB + 1), dim3(256), 0, stream>>>(
        query, Wq, Wloc, conv_w, qproj_ws, wc_ws);

    score_kernel<<<dim3(TT / TILE_T, BB), dim3(256), 0, stream>>>(
        keys, prev_al, mask, v_w, v_b, qproj_ws, wc_ws, align_out);

    softmax_ctx_kernel<<<dim3(BB), dim3(1024), 0, stream>>>(
        values, align_out, att_out);
}